// upFirDn2D_50465865728269
// MI455X (gfx1250) — compile-verified
//
#include <hip/hip_runtime.h>

typedef float v2f __attribute__((ext_vector_type(2)));
typedef float v8f __attribute__((ext_vector_type(8)));
typedef float v4f __attribute__((ext_vector_type(4)));

#define IH 128
#define IW 128
#define OH 127
#define OW 127
#define LDS_STRIDE 136   // payload at [4..131]; 544B row pitch keeps 16B alignment

// Depthwise separable [1,3,3,1]x[1,3,3,1]/64 blur.
// Staging: GLOBAL_LOAD_ASYNC_TO_LDS_B128 (512B per wave-row, no VGPR roundtrip).
// Horizontal pass: VALU (v_pk_fma_f32) from LDS.
// Vertical pass: V_WMMA_F32_16X16X4_F32, Out(16x16) = Fv(16x20)*T(20x16), 5 K-chunks.
__global__ __launch_bounds__(256)
void upfirdn2d_blur_kernel(const float* __restrict__ x, float* __restrict__ out) {
    __shared__ __align__(16) float X[20][LDS_STRIDE];

    const int tid   = threadIdx.x;
    const int plane = blockIdx.x >> 3;   // n*C + c, 4096 planes
    const int strip = blockIdx.x & 7;    // 8 strips of 16 output rows
    const int i0    = strip << 4;

    const float* xp = x + (size_t)plane * (IH * IW);

    const int wave = tid >> 5;
    const int lane = tid & 31;

    // ---- Stage rows i0-1 .. i0+18 (20 rows): one async B128 per lane per row ----
    // Global col 4*lane..4*lane+3 -> X[r][4 + 4*lane ..]; wave w owns rows w, w+8, w+16.
    {
        const int lbyte_col = (4 + 4 * lane) * 4;          // LDS byte offset within row
        const int gbyte_col = (4 * lane) * 4;              // global byte offset within row
        for (int r = wave; r < 20; r += 8) {
            const int grow = i0 - 1 + r;
            const unsigned lds_off =
                (unsigned)(size_t)(&X[0][0]) + (unsigned)(r * LDS_STRIDE * 4 + lbyte_col);
            if (grow >= 0 && grow < IH) {
                const unsigned g_off = (unsigned)(grow * IW * 4 + gbyte_col);
                asm volatile("global_load_async_to_lds_b128 %0, %1, %2"
                             :: "v"(lds_off), "v"(g_off), "s"(xp)
                             : "memory");
            } else {
                v4f z = {};
                *(v4f*)((char*)&X[0][0] + (r * LDS_STRIDE * 4 + lbyte_col)) = z;
            }
        }
        if (tid < 20) {   // zero halos: global col -1 (idx 3), cols 128,129 (idx 132,133)
            X[tid][3]   = 0.0f;
            X[tid][132] = 0.0f;
            X[tid][133] = 0.0f;
        }
    }
    asm volatile("s_wait_asynccnt 0x0" ::: "memory");
    __syncthreads();

    const int lo = lane & 15;
    const int hi = lane >> 4;
    const int j0 = wave << 4;

    // ---- A operand: vertical band filter Fv[m,k] = f[k-m]/64, f={1,3,3,1} ----
    // A 16x4 layout: lanes0-15 hold K={0,1} (VGPR0,1), lanes16-31 K={2,3}.
    v2f A[5];
    #pragma unroll
    for (int c = 0; c < 5; ++c) {
        #pragma unroll
        for (int s = 0; s < 2; ++s) {
            const int d = (4 * c + 2 * hi + s) - lo;
            float fv = 0.0f;
            if (d == 0 || d == 3) fv = 1.0f / 64.0f;
            else if (d == 1 || d == 2) fv = 3.0f / 64.0f;
            if (s == 0) A[c].x = fv; else A[c].y = fv;
        }
    }

    // ---- Horizontal taps feed B operands; WMMA accumulates vertical pass ----
    // B 4x16 layout: lanes0-15 hold K={0,1}, lanes16-31 K={2,3};
    // chunk c: this lane supplies T[4c+2hi, lo] and T[4c+2hi+1, lo].
    v8f acc = {};
    const int base = 3 + j0 + lo;   // index of global col (j0+lo-1)
    #pragma unroll
    for (int c = 0; c < 5; ++c) {
        const int ra = 4 * c + 2 * hi;
        const float* r0 = &X[ra][base];
        const float* r1 = &X[ra + 1][base];
        v2f B;
        B.x = (r0[0] + r0[3]) + 3.0f * (r0[1] + r0[2]);
        B.y = (r1[0] + r1[3]) + 3.0f * (r1[1] + r1[2]);
        acc = __builtin_amdgcn_wmma_f32_16x16x4_f32(
            /*neg_a=*/false, A[c], /*neg_b=*/false, B,
            /*c_mod=*/(short)0, acc, /*reuse_a=*/false, /*reuse_b=*/false);
    }

    // ---- Store: C/D layout -> acc[p] = Out[p + 8*hi, lo] ----
    const int ocol = j0 + lo;
    if (ocol < OW) {
        float* op = out + (size_t)plane * (OH * OW);
        #pragma unroll
        for (int p = 0; p < 8; ++p) {
            const int orow = i0 + p + 8 * hi;
            if (orow < OH) op[orow * OW + ocol] = acc[p];
        }
    }
}

extern "C" void kernel_launch(void* const* d_in, const int* in_sizes, int n_in,
                              void* d_out, int out_size, void* d_ws, size_t ws_size,
                              hipStream_t stream) {
    const float* x = (const float*)d_in[0];
    float* out = (float*)d_out;
    (void)in_sizes; (void)n_in; (void)out_size; (void)d_ws; (void)ws_size;

    const int planes = 16 * 256;           // N * C
    dim3 grid(planes * 8);                 // 8 row-strips per plane
    dim3 block(256);                       // 8 waves; 1 wave per 16x16 output tile
    upfirdn2d_blur_kernel<<<grid, block, 0, stream>>>(x, out);
}